// Multi_head_Attention_55594056680006
// MI455X (gfx1250) — compile-verified
//
#include <hip/hip_runtime.h>

typedef __attribute__((ext_vector_type(2))) float v2f;
typedef __attribute__((ext_vector_type(8))) float v8f;

#define TILE_M 64
#define TILE_N 64
#define TILE_K 32
#define LDS_P  (TILE_K + 4)   // 36-float row stride: 144B, 16B-aligned, bank-spread

// Generic fp32 GEMM on the CDNA5 f32 matrix core:
//   C[M,N] = alpha * A[M,K] @ (transB ? B[N,K]^T : B[K,N]) + bias[N]
// Block = 256 threads (8 wave32), 64x64 C tile, K tiled by 32 through LDS.
// Both A and B tiles are stored K-contiguous in LDS so every WMMA source
// fragment is one aligned ds_load_b64 into an even VGPR pair.
// Global->register prefetch of the next tile overlaps HBM latency with WMMA.
__global__ __launch_bounds__(256) void wmma_gemm_f32(
    const float* __restrict__ A, const float* __restrict__ B,
    const float* __restrict__ bias, float* __restrict__ C,
    int M, int N, int K, int lda, int ldb, int ldc,
    float alpha, int transB)
{
  __shared__ float As[TILE_M][LDS_P];   // [m][k]
  __shared__ float Bs[TILE_N][LDS_P];   // [n][k]  (K-contiguous!)

  const int tid  = threadIdx.x;
  const int lane = tid & 31;
  const int wid  = tid >> 5;
  const int h    = lane >> 4;    // lane half (ISA fragment layout selector)
  const int ln   = lane & 15;

  const int mBase = blockIdx.y * TILE_M;
  const int nBase = blockIdx.x * TILE_N;

  const int m0 = (wid & 3) * 16;   // wave's M sub-tile
  const int n0 = (wid >> 2) * 32;  // wave's N sub-tile (two 16-wide halves)

  // staging thread coordinates
  const int arow = tid >> 3;          // 0..31 (+32 for second quad)
  const int ac4  = (tid & 7) * 4;
  const int bkr  = tid >> 4;          // 0..15 (+16): K row, non-transB
  const int bnc  = (tid & 15) * 4;    // N col quad, non-transB
  const int tn   = tid >> 3;          // 0..31 (+32): N row, transB
  const int tc4  = (tid & 7) * 4;     // K col quad, transB

  float4 pa0, pa1, pb0, pb1;

  auto gload = [&](int k0) {
    pa0 = *reinterpret_cast<const float4*>(
        &A[(size_t)(mBase + arow) * lda + (size_t)(k0 + ac4)]);
    pa1 = *reinterpret_cast<const float4*>(
        &A[(size_t)(mBase + arow + 32) * lda + (size_t)(k0 + ac4)]);
    if (!transB) {
      pb0 = *reinterpret_cast<const float4*>(
          &B[(size_t)(k0 + bkr) * ldb + (size_t)(nBase + bnc)]);
      pb1 = *reinterpret_cast<const float4*>(
          &B[(size_t)(k0 + bkr + 16) * ldb + (size_t)(nBase + bnc)]);
    } else {
      pb0 = *reinterpret_cast<const float4*>(
          &B[(size_t)(nBase + tn) * ldb + (size_t)(k0 + tc4)]);
      pb1 = *reinterpret_cast<const float4*>(
          &B[(size_t)(nBase + tn + 32) * ldb + (size_t)(k0 + tc4)]);
    }
  };

  auto sstore = [&]() {
    *reinterpret_cast<float4*>(&As[arow][ac4])      = pa0;
    *reinterpret_cast<float4*>(&As[arow + 32][ac4]) = pa1;
    if (!transB) {
      // transpose [k][n] -> Bs[n][k] (once per tile; keeps hot loop clean)
      Bs[bnc + 0][bkr] = pb0.x;  Bs[bnc + 1][bkr] = pb0.y;
      Bs[bnc + 2][bkr] = pb0.z;  Bs[bnc + 3][bkr] = pb0.w;
      Bs[bnc + 0][bkr + 16] = pb1.x;  Bs[bnc + 1][bkr + 16] = pb1.y;
      Bs[bnc + 2][bkr + 16] = pb1.z;  Bs[bnc + 3][bkr + 16] = pb1.w;
    } else {
      *reinterpret_cast<float4*>(&Bs[tn][tc4])      = pb0;
      *reinterpret_cast<float4*>(&Bs[tn + 32][tc4]) = pb1;
    }
  };

  v8f acc0 = {0.f, 0.f, 0.f, 0.f, 0.f, 0.f, 0.f, 0.f};
  v8f acc1 = {0.f, 0.f, 0.f, 0.f, 0.f, 0.f, 0.f, 0.f};

  gload(0);  // prime the pipeline

  for (int k0 = 0; k0 < K; k0 += TILE_K) {
    sstore();
    __syncthreads();

    if (k0 + TILE_K < K) gload(k0 + TILE_K);  // in flight during the WMMAs

    // ---- 8 K-steps of V_WMMA_F32_16X16X4_F32, two N halves each ----
#pragma unroll
    for (int kk = 0; kk < TILE_K; kk += 4) {
      const int kb = kk + 2 * h;           // ISA A/B layout: K = vgpr + 2*laneHalf
      v2f a  = *reinterpret_cast<const v2f*>(&As[m0 + ln][kb]);
      v2f b0 = *reinterpret_cast<const v2f*>(&Bs[n0 + ln][kb]);
      v2f b1 = *reinterpret_cast<const v2f*>(&Bs[n0 + 16 + ln][kb]);
      acc0 = __builtin_amdgcn_wmma_f32_16x16x4_f32(
          false, a, false, b0, (short)0, acc0, false, false);
      acc1 = __builtin_amdgcn_wmma_f32_16x16x4_f32(
          false, a, false, b1, (short)0, acc1, false, false);
    }
    __syncthreads();
  }

  // ---- epilogue: scale + bias, ISA C/D layout M = r + 8*laneHalf ----
#pragma unroll
  for (int r = 0; r < 8; ++r) {
    int gm  = mBase + m0 + r + 8 * h;
    int gn0 = nBase + n0 + ln;
    int gn1 = gn0 + 16;
    float o0 = acc0[r] * alpha;
    float o1 = acc1[r] * alpha;
    if (bias) { o0 += bias[gn0]; o1 += bias[gn1]; }
    C[(size_t)gm * ldc + gn0] = o0;
    C[(size_t)gm * ldc + gn1] = o1;
  }
}

// In-place row softmax: one 256-thread block per row.
__global__ __launch_bounds__(256) void softmax_rows(float* __restrict__ attn, int n)
{
  __shared__ float red[256];
  const int tid = threadIdx.x;
  float* p = attn + (size_t)blockIdx.x * n;

  float m = -__builtin_inff();
  for (int i = tid; i < n; i += 256) m = fmaxf(m, p[i]);
  red[tid] = m;
  __syncthreads();
  for (int s = 128; s > 0; s >>= 1) {
    if (tid < s) red[tid] = fmaxf(red[tid], red[tid + s]);
    __syncthreads();
  }
  m = red[0];
  __syncthreads();

  float sum = 0.f;
  for (int i = tid; i < n; i += 256) {
    float e = expf(p[i] - m);
    p[i] = e;
    sum += e;
  }
  red[tid] = sum;
  __syncthreads();
  for (int s = 128; s > 0; s >>= 1) {
    if (tid < s) red[tid] += red[tid + s];
    __syncthreads();
  }
  float inv = 1.0f / red[0];
  for (int i = tid; i < n; i += 256) p[i] *= inv;
}

extern "C" void kernel_launch(void* const* d_in, const int* in_sizes, int n_in,
                              void* d_out, int out_size, void* d_ws, size_t ws_size,
                              hipStream_t stream)
{
  (void)in_sizes; (void)n_in; (void)out_size; (void)ws_size;

  const float* x    = (const float*)d_in[0];  // [8192, 512]
  const float* Amat = (const float*)d_in[1];  // [8192, 8192]
  const float* Wqkv = (const float*)d_in[2];  // [512, 1536]
  const float* bqkv = (const float*)d_in[3];  // [1536]
  const float* Wo   = (const float*)d_in[4];  // [512, 512]
  const float* bo   = (const float*)d_in[5];  // [512]

  const int N = 8192, D = 512, D3 = 1536;
  const float scale = 0.044194173824159216f;  // 1/sqrt(512)

  float* out_o = (float*)d_out;               // [N, D]
  float* attn  = out_o + (size_t)N * D;       // [N, N] (output #2)

  float* qkv = (float*)d_ws;                  // [N, 1536]   48 MB
  float* k2  = qkv + (size_t)N * D3;          // [N, 512]    16 MB
  float* val = k2 + (size_t)N * D;            // [N, 512]    16 MB

  dim3 blk(256);

  // 1) qkv = x @ Wqkv + bqkv
  wmma_gemm_f32<<<dim3(D3 / 64, N / 64), blk, 0, stream>>>(
      x, Wqkv, bqkv, qkv, N, D3, D, D, D3, D3, 1.0f, 0);

  // 2) k2 = A @ k   (k = qkv[:, 512:1024], ldb = 1536)
  wmma_gemm_f32<<<dim3(D / 64, N / 64), blk, 0, stream>>>(
      Amat, qkv + D, nullptr, k2, N, D, N, N, D3, D, 1.0f, 0);

  // 3) attn = (q @ k2^T) * scale   (q = qkv[:, 0:512], lda = 1536; B transposed)
  wmma_gemm_f32<<<dim3(N / 64, N / 64), blk, 0, stream>>>(
      qkv, k2, nullptr, attn, N, N, D, D3, D, N, scale, 1);

  // 4) softmax rows, in place in d_out
  softmax_rows<<<dim3(N), blk, 0, stream>>>(attn, N);

  // 5) val = attn @ v   (v = qkv[:, 1024:1536], ldb = 1536)
  wmma_gemm_f32<<<dim3(D / 64, N / 64), blk, 0, stream>>>(
      attn, qkv + 2 * D, nullptr, val, N, D, N, N, D3, D, 1.0f, 0);

  // 6) o = val @ Wo + bo
  wmma_gemm_f32<<<dim3(D / 64, N / 64), blk, 0, stream>>>(
      val, Wo, bo, out_o, N, D, D, D, D, D, 1.0f, 0);
}